// GNNEncoder_35845797053073
// MI455X (gfx1250) — compile-verified
//
#include <hip/hip_runtime.h>

typedef float v2f __attribute__((ext_vector_type(2)));
typedef float v8f __attribute__((ext_vector_type(8)));

#define DHID 256

// ---------- CDNA5 async global->LDS helpers ----------
__device__ __forceinline__ unsigned to_lds_addr(const void* p) {
  // generic -> LDS(as3) pointer, then to 32-bit LDS byte address
  return (unsigned)(unsigned long long)(const __attribute__((address_space(3))) void*)p;
}

__device__ __forceinline__ void async_copy_b128(unsigned lds_dst, const void* gsrc) {
  asm volatile("global_load_async_to_lds_b128 %0, %1, off"
               :
               : "v"(lds_dst), "v"((unsigned long long)gsrc)
               : "memory");
}

__device__ __forceinline__ void wait_async0() {
  asm volatile("s_wait_asynccnt 0x0" ::: "memory");
}

// ---------------- degree / norm ----------------
__global__ __launch_bounds__(256) void k_init_deg(float* __restrict__ deg, int n) {
  int i = blockIdx.x * blockDim.x + threadIdx.x;
  if (i < n) deg[i] = 1.0f;  // self-loop
}

__global__ __launch_bounds__(256) void k_deg_accum(const int* __restrict__ dst,
                                                   float* __restrict__ deg, int ne) {
  int i = blockIdx.x * blockDim.x + threadIdx.x;
  if (i < ne) atomicAdd(&deg[dst[i]], 1.0f);
}

__global__ __launch_bounds__(256) void k_rsqrt(float* __restrict__ deg, int n) {
  int i = blockIdx.x * blockDim.x + threadIdx.x;
  if (i < n) deg[i] = rsqrtf(deg[i]);
}

// ---------------- GEMM: Out[n,256] = X[n,256] @ W[256,256] via V_WMMA_F32_16X16X4_F32 ----
// Block: 256 threads = 8 waves. Block tile = 32 rows x 256 cols.
// Waves 0-3: rows [0,16), col quadrants 0..3 (64 cols each). Waves 4-7: rows [16,32).
// Both W (32x256, 32KB) and A (32x32, 4KB) slabs staged via async global->LDS copies.
__global__ __launch_bounds__(256) void k_gemm_wmma(const float* __restrict__ X,
                                                   const float* __restrict__ W,
                                                   float* __restrict__ Out,
                                                   int nrows) {
  __shared__ float wlds[32 * DHID];  // [k][n]
  __shared__ float alds[32 * 32];    // [row][k]

  const int tid   = threadIdx.x;
  const int wave  = tid >> 5;
  const int lane  = tid & 31;
  const int lhalf = lane >> 4;   // 0: lanes 0-15, 1: lanes 16-31
  const int l15   = lane & 15;
  const int row0  = blockIdx.x * 32 + (wave >> 2) * 16;
  const int col0  = (wave & 3) * 64;

  v8f acc0 = {}; v8f acc1 = {}; v8f acc2 = {}; v8f acc3 = {};

  // per-thread async-copy source/dest for the A slab: row = blk*32 + tid/8, 16B chunk = tid%8
  int ar = blockIdx.x * 32 + (tid >> 3);
  if (ar >= nrows) ar = nrows - 1;  // clamp: garbage values only feed store-guarded rows
  const float* asrc    = X + (size_t)ar * DHID + (tid & 7) * 4;
  const unsigned adst  = to_lds_addr(&alds[(tid >> 3) * 32 + (tid & 7) * 4]);
  const unsigned wbase = to_lds_addr(wlds);
  const int      aoff  = ((wave >> 2) * 16 + l15) * 32;  // this lane's A row in alds

  for (int kb = 0; kb < DHID; kb += 32) {
    __syncthreads();  // previous slab fully consumed before overwrite
    // W slab: 32x256 f32 = 2048 16B-chunks, 8 per thread, all in flight
#pragma unroll
    for (int i = 0; i < 8; ++i) {
      const int idx = tid + i * 256;  // float4 index within slab
      async_copy_b128(wbase + (unsigned)idx * 16, W + (size_t)kb * DHID + (size_t)idx * 4);
    }
    // A slab: one 16B chunk per thread
    async_copy_b128(adst, asrc + kb);
    wait_async0();
    __syncthreads();

#pragma unroll
    for (int k = 0; k < 32; k += 4) {
      const int kk = k + lhalf * 2;  // this lane's K pair within the slab
      // A fragment: 16x4 f32 (lanes 0-15 -> K=k,k+1 ; lanes 16-31 -> K=k+2,k+3)
      v2f a = *(const v2f*)&alds[aoff + kk];
      // B fragments: 4x16 f32 per col tile, same K split across lane halves
      const float* b0p = wlds + kk * DHID + col0 + l15;
      const float* b1p = b0p + DHID;
      v2f b0; b0.x = b0p[0];  b0.y = b1p[0];
      v2f b1; b1.x = b0p[16]; b1.y = b1p[16];
      v2f b2; b2.x = b0p[32]; b2.y = b1p[32];
      v2f b3; b3.x = b0p[48]; b3.y = b1p[48];

      acc0 = __builtin_amdgcn_wmma_f32_16x16x4_f32(false, a, false, b0, (short)0, acc0, false, false);
      acc1 = __builtin_amdgcn_wmma_f32_16x16x4_f32(false, a, false, b1, (short)0, acc1, false, false);
      acc2 = __builtin_amdgcn_wmma_f32_16x16x4_f32(false, a, false, b2, (short)0, acc2, false, false);
      acc3 = __builtin_amdgcn_wmma_f32_16x16x4_f32(false, a, false, b3, (short)0, acc3, false, false);
    }
  }

  // C/D layout: VGPR r -> M = r (lanes 0-15) or M = 8+r (lanes 16-31); N = lane&15
#pragma unroll
  for (int r = 0; r < 8; ++r) {
    const int m = row0 + lhalf * 8 + r;
    if (m < nrows) {
      float* op = Out + (size_t)m * DHID + col0 + l15;
      op[0]  = acc0[r];
      op[16] = acc1[r];
      op[32] = acc2[r];
      op[48] = acc3[r];
    }
  }
}

// ---------------- self-loop init: out = t * dinv[node]^2 (initializes accumulator) ----
__global__ __launch_bounds__(256) void k_self_init(const float* __restrict__ t,
                                                   const float* __restrict__ dinv,
                                                   float* __restrict__ out, int n) {
  int idx = blockIdx.x * blockDim.x + threadIdx.x;  // over n*64 float4
  int total = n * (DHID / 4);
  if (idx >= total) return;
  int node = idx >> 6;
  float s = dinv[node];
  s = s * s;
  float4 v = ((const float4*)t)[idx];
  float4 o;
  o.x = v.x * s; o.y = v.y * s; o.z = v.z * s; o.w = v.w * s;
  ((float4*)out)[idx] = o;
}

// ---------------- edge scatter: out[dst] += t[src] * dinv[src]*dinv[dst] (one wave / edge) --
__global__ __launch_bounds__(256) void k_scatter(const float* __restrict__ t,
                                                 const int* __restrict__ src,
                                                 const int* __restrict__ dst,
                                                 const float* __restrict__ dinv,
                                                 float* __restrict__ out, int ne) {
  int e = blockIdx.x * (blockDim.x >> 5) + (threadIdx.x >> 5);
  if (e >= ne) return;
  int lane = threadIdx.x & 31;
  int s = src[e], d = dst[e];
  float norm = dinv[s] * dinv[d];
  const float4* tp = (const float4*)(t + (size_t)s * DHID) + lane * 2;
  float4 v0 = tp[0], v1 = tp[1];
  float* op = out + (size_t)d * DHID + lane * 8;
  atomicAdd(op + 0, v0.x * norm);
  atomicAdd(op + 1, v0.y * norm);
  atomicAdd(op + 2, v0.z * norm);
  atomicAdd(op + 3, v0.w * norm);
  atomicAdd(op + 4, v1.x * norm);
  atomicAdd(op + 5, v1.y * norm);
  atomicAdd(op + 6, v1.z * norm);
  atomicAdd(op + 7, v1.w * norm);
}

// ---------------- bias + relu ----------------
__global__ __launch_bounds__(256) void k_bias_relu(float* __restrict__ out,
                                                   const float* __restrict__ bias, int n) {
  int idx = blockIdx.x * blockDim.x + threadIdx.x;  // over n*64 float4
  int total = n * (DHID / 4);
  if (idx >= total) return;
  int j4 = (idx & 63) * 4;
  float4 v = ((float4*)out)[idx];
  v.x = fmaxf(v.x + bias[j4 + 0], 0.0f);
  v.y = fmaxf(v.y + bias[j4 + 1], 0.0f);
  v.z = fmaxf(v.z + bias[j4 + 2], 0.0f);
  v.w = fmaxf(v.w + bias[j4 + 3], 0.0f);
  ((float4*)out)[idx] = v;
}

// ---------------- edge predictor: y = ((h[s]+h[d])*0.5) . We + be (one wave / edge) ----
__global__ __launch_bounds__(256) void k_edge_pred(const float* __restrict__ h,
                                                   const int* __restrict__ e0,
                                                   const int* __restrict__ e1,
                                                   const float* __restrict__ We,
                                                   const float* __restrict__ be,
                                                   float* __restrict__ y, int ne) {
  int e = blockIdx.x * (blockDim.x >> 5) + (threadIdx.x >> 5);
  if (e >= ne) return;
  int lane = threadIdx.x & 31;
  int a = e0[e], b = e1[e];
  const float4* pa = (const float4*)(h + (size_t)a * DHID) + lane * 2;
  const float4* pb = (const float4*)(h + (size_t)b * DHID) + lane * 2;
  const float4* pw = (const float4*)We + lane * 2;
  float4 a0 = pa[0], a1 = pa[1];
  float4 b0 = pb[0], b1 = pb[1];
  float4 w0 = pw[0], w1 = pw[1];
  float s = (a0.x + b0.x) * 0.5f * w0.x + (a0.y + b0.y) * 0.5f * w0.y +
            (a0.z + b0.z) * 0.5f * w0.z + (a0.w + b0.w) * 0.5f * w0.w +
            (a1.x + b1.x) * 0.5f * w1.x + (a1.y + b1.y) * 0.5f * w1.y +
            (a1.z + b1.z) * 0.5f * w1.z + (a1.w + b1.w) * 0.5f * w1.w;
#pragma unroll
  for (int off = 16; off > 0; off >>= 1) s += __shfl_xor(s, off, 32);
  if (lane == 0) y[e] = s + be[0];
}

extern "C" void kernel_launch(void* const* d_in, const int* in_sizes, int n_in,
                              void* d_out, int out_size, void* d_ws, size_t ws_size,
                              hipStream_t stream) {
  const float* x  = (const float*)d_in[0];
  const int*   ei = (const int*)d_in[1];
  const float* W1 = (const float*)d_in[2];
  const float* b1 = (const float*)d_in[3];
  const float* W2 = (const float*)d_in[4];
  const float* b2 = (const float*)d_in[5];
  const float* We = (const float*)d_in[6];
  const float* be = (const float*)d_in[7];

  const int n = in_sizes[0] / DHID;  // 50000
  const int e = in_sizes[1] / 2;     // 800000
  const int* src = ei;
  const int* dst = ei + e;

  // workspace: dinv [n], then t [n*256] (aligned)
  size_t dinvBytes = (((size_t)n * sizeof(float)) + 1023) & ~(size_t)1023;
  float* dinv = (float*)d_ws;
  float* t    = (float*)((char*)d_ws + dinvBytes);

  float* h = (float*)d_out;                     // [n, 256]
  float* y = (float*)d_out + (size_t)n * DHID;  // [e]

  const int tot4 = n * (DHID / 4);

  // degree / normalization
  k_init_deg <<<(n + 255) / 256, 256, 0, stream>>>(dinv, n);
  k_deg_accum<<<(e + 255) / 256, 256, 0, stream>>>(dst, dinv, e);
  k_rsqrt    <<<(n + 255) / 256, 256, 0, stream>>>(dinv, n);

  // layer 1: t = x @ W1 ; h = aggregate ; relu(h + b1)
  k_gemm_wmma<<<(n + 31) / 32, 256, 0, stream>>>(x, W1, t, n);
  k_self_init<<<(tot4 + 255) / 256, 256, 0, stream>>>(t, dinv, h, n);
  k_scatter  <<<(e + 7) / 8, 256, 0, stream>>>(t, src, dst, dinv, h, e);
  k_bias_relu<<<(tot4 + 255) / 256, 256, 0, stream>>>(h, b1, n);

  // layer 2: t = h @ W2 ; h = aggregate ; relu(h + b2)
  k_gemm_wmma<<<(n + 31) / 32, 256, 0, stream>>>(h, W2, t, n);
  k_self_init<<<(tot4 + 255) / 256, 256, 0, stream>>>(t, dinv, h, n);
  k_scatter  <<<(e + 7) / 8, 256, 0, stream>>>(t, src, dst, dinv, h, e);
  k_bias_relu<<<(tot4 + 255) / 256, 256, 0, stream>>>(h, b2, n);

  // edge predictor on original endpoints
  k_edge_pred<<<(e + 7) / 8, 256, 0, stream>>>(h, src, dst, We, be, y, e);
}